// MultiScaleRoIAlign_35072702939760
// MI455X (gfx1250) — compile-verified
//
#include <hip/hip_runtime.h>
#include <hip/hip_bf16.h>
#include <cstdint>

// ---------------------------------------------------------------------------
// Problem constants (match reference setup_inputs exactly)
// ---------------------------------------------------------------------------
#define BATCH   2
#define NBOX    512
#define KROI    (BATCH * NBOX)     // 1024
#define CCH     256
#define OUTHW   7
#define NBIN    (OUTHW * OUTHW)    // 49
#define OUT_PER_ROI (CCH * NBIN)   // 12544

__device__ __constant__ const int   kHs[4]  = {200, 100, 50, 25};
__device__ __constant__ const float kScl[4] = {0.25f, 0.125f, 0.0625f, 0.03125f};

// NHWC level offsets in the workspace (floats)
#define L0_OFF 0L
#define L1_OFF 20480000L            // + 2*200*200*256
#define L2_OFF 25600000L            // + 2*100*100*256
#define L3_OFF 26880000L            // + 2*50*50*256
#define NHWC_TOTAL_FLOATS 27200000L // + 2*25*25*256

// ---------------------------------------------------------------------------
// Kernel 1: NCHW -> NHWC transpose, one wave32 per 32x32 (C x HW) tile.
// Strided source rows are pulled into LDS with the CDNA5 async-to-LDS path
// (global_load_async_to_lds_b128 + s_wait_asynccnt), then written out with
// lanes contiguous along C (128B coalesced stores).
// ---------------------------------------------------------------------------
__global__ __launch_bounds__(32) void nchw_to_nhwc(const float* __restrict__ src,
                                                   float* __restrict__ dst,
                                                   int HW) {
    const unsigned lane = threadIdx.x;          // 0..31
    const int hw0 = blockIdx.x * 32;
    const int c0  = blockIdx.y * 32;
    const size_t b = blockIdx.z;

    const float* srcb = src + (b * CCH + (size_t)c0) * (size_t)HW;
    float*       dstb = dst + (b * (size_t)HW) * CCH;

    __shared__ __align__(16) float tile[32 * 36];   // 36-float row pitch: 16B aligned, 2-way bank conflict max

    const bool asyncOK = ((HW & 3) == 0) && (hw0 + 32 <= HW);
    if (asyncOK) {
        // 8 async wave-instructions: 32 lanes x 16B = 512B each -> 4KB tile
        const unsigned chunk = lane & 7u;       // 16B chunk within a 128B row
        const unsigned rbase = lane >> 3;       // 0..3
        const unsigned ldsBase = (unsigned)(size_t)(&tile[0]);
        const unsigned long long sbase = (unsigned long long)(size_t)srcb;
#pragma unroll
        for (unsigned i = 0; i < 8; ++i) {
            const unsigned row  = i * 4u + rbase;
            const unsigned gofs = (row * (unsigned)HW + (unsigned)hw0 + chunk * 4u) * 4u;
            const unsigned lofs = ldsBase + (row * 36u + chunk * 4u) * 4u;
            asm volatile("global_load_async_to_lds_b128 %0, %1, %2"
                         :: "v"(lofs), "v"(gofs), "s"(sbase)
                         : "memory");
        }
        asm volatile("s_wait_asynccnt 0" ::: "memory");
    } else {
        // Ragged / unaligned tail: synchronous, clamped loads (still coalesced rows)
        int hw = hw0 + (int)lane;
        if (hw >= HW) hw = HW - 1;
#pragma unroll 4
        for (int r = 0; r < 32; ++r)
            tile[r * 36 + lane] = srcb[(size_t)r * HW + hw];
    }

    // Store transposed: lane = channel, contiguous 128B per wave store
#pragma unroll 4
    for (int r = 0; r < 32; ++r) {
        const int hw = hw0 + r;
        if (hw < HW)
            dstb[(size_t)hw * CCH + c0 + lane] = tile[lane * 36 + r];
    }
}

// ---------------------------------------------------------------------------
// Kernel 2: ROI align. One 256-thread block (8 waves) per ROI.
//  phase 0: 196 threads compute sample geometry (tap base + 4 bilinear weights)
//  phase 1: thread = (c4 group of 4 channels, bin row); float4 tap gathers
//  phase 2: coalesced write of the 12544-float ROI result from LDS
// ---------------------------------------------------------------------------
__global__ __launch_bounds__(256) void roi_align(const float* __restrict__ f0,
                                                 const float* __restrict__ f1,
                                                 const float* __restrict__ f2,
                                                 const float* __restrict__ f3,
                                                 const float* __restrict__ boxes,
                                                 float* __restrict__ outp,
                                                 const float* __restrict__ nhwc,
                                                 int useNHWC) {
    const int roi = blockIdx.x;
    const int t   = threadIdx.x;
    const int b   = roi >> 9;            // N = 512

    // ---- per-ROI scalars (uniform, computed by every thread; cheap) ----
    const float bx1 = boxes[roi * 4 + 0];
    const float by1 = boxes[roi * 4 + 1];
    const float bx2 = boxes[roi * 4 + 2];
    const float by2 = boxes[roi * 4 + 3];
    const float area = fmaxf((bx2 - bx1) * (by2 - by1), 0.0f);
    const float sedge = sqrtf(area);
    float lv = (sedge > 0.0f) ? floorf(4.0f + log2f(sedge / 224.0f) + 1e-6f) : 2.0f;
    const int L = (int)fminf(fmaxf(lv, 2.0f), 5.0f) - 2;

    const int   H = kHs[L];
    const int   W = H;
    const float scale = kScl[L];
    const float x1 = bx1 * scale, y1 = by1 * scale;
    const float x2 = bx2 * scale, y2 = by2 * scale;
    const float roiw = fmaxf(x2 - x1, 1.0f);
    const float roih = fmaxf(y2 - y1, 1.0f);
    const float binw = roiw * (1.0f / OUTHW);
    const float binh = roih * (1.0f / OUTHW);

    __shared__ int   gspat[NBIN * 4];
    __shared__ float gw[NBIN * 4 * 4];
    __shared__ float st[OUT_PER_ROI];        // 50176 B staging for coalesced output

    // ---- phase 0: geometry for 49 bins x 4 subsamples ----
    if (t < NBIN * 4) {
        const int bin = t >> 2, sub = t & 3;
        const int ph = bin / OUTHW, pw = bin % OUTHW;
        const int sy = sub >> 1, sx = sub & 1;
        const float y = y1 + ph * binh + (sy + 0.5f) * binh * 0.5f;
        const float x = x1 + pw * binw + (sx + 0.5f) * binw * 0.5f;
        const bool valid = (y >= -1.0f) && (y <= (float)H) && (x >= -1.0f) && (x <= (float)W);
        const float yc = fminf(fmaxf(y, 0.0f), (float)(H - 1));
        const float xc = fminf(fmaxf(x, 0.0f), (float)(W - 1));
        int yl = (int)floorf(yc); yl = yl < 0 ? 0 : (yl > H - 2 ? H - 2 : yl);
        int xl = (int)floorf(xc); xl = xl < 0 ? 0 : (xl > W - 2 ? W - 2 : xl);
        const float ly = yc - (float)yl, lx = xc - (float)xl;
        float w00 = (1.0f - ly) * (1.0f - lx);
        float w01 = (1.0f - ly) * lx;
        float w10 = ly * (1.0f - lx);
        float w11 = ly * lx;
        if (!valid) { w00 = w01 = w10 = w11 = 0.0f; }
        gspat[t] = yl * W + xl;
        gw[t * 4 + 0] = w00; gw[t * 4 + 1] = w01;
        gw[t * 4 + 2] = w10; gw[t * 4 + 3] = w11;
    }
    __syncthreads();

    // ---- phase 1: gather + bilinear combine ----
    const int c4   = t & 63;     // group of 4 channels (float4)
    const int brow = t >> 6;     // 0..3 -> bins brow, brow+4, ...
    const long lvlOff[4] = {L0_OFF, L1_OFF, L2_OFF, L3_OFF};

    if (useNHWC) {
        const float* base = nhwc + lvlOff[L] + ((long)b * H * W) * CCH + (long)c4 * 4;
        const long rs = (long)W * CCH;   // +1 row in y
        for (int i = 0; i < 13; ++i) {
            const int bin = brow + 4 * i;
            if (bin >= NBIN) break;
            // Prefetch the next bin's first tap vector (global_prefetch_b8)
            const int nb = bin + 4;
            if (nb < NBIN)
                __builtin_prefetch(base + (long)gspat[nb * 4] * CCH, 0, 1);
            float4 acc = {0.0f, 0.0f, 0.0f, 0.0f};
#pragma unroll
            for (int sub = 0; sub < 4; ++sub) {
                const int g = bin * 4 + sub;
                const long sp = (long)gspat[g] * CCH;
                const float w0 = gw[g * 4 + 0], w1 = gw[g * 4 + 1];
                const float w2 = gw[g * 4 + 2], w3 = gw[g * 4 + 3];
                const float4 v00 = *(const float4*)(base + sp);
                const float4 v01 = *(const float4*)(base + sp + CCH);
                const float4 v10 = *(const float4*)(base + sp + rs);
                const float4 v11 = *(const float4*)(base + sp + rs + CCH);
                acc.x += w0 * v00.x + w1 * v01.x + w2 * v10.x + w3 * v11.x;
                acc.y += w0 * v00.y + w1 * v01.y + w2 * v10.y + w3 * v11.y;
                acc.z += w0 * v00.z + w1 * v01.z + w2 * v10.z + w3 * v11.z;
                acc.w += w0 * v00.w + w1 * v01.w + w2 * v10.w + w3 * v11.w;
            }
            const int c = c4 * 4;
            st[(c + 0) * NBIN + bin] = acc.x * 0.25f;
            st[(c + 1) * NBIN + bin] = acc.y * 0.25f;
            st[(c + 2) * NBIN + bin] = acc.z * 0.25f;
            st[(c + 3) * NBIN + bin] = acc.w * 0.25f;
        }
    } else {
        // NCHW fallback (workspace too small): correct but strided
        const float* fsrc = (L == 0) ? f0 : (L == 1) ? f1 : (L == 2) ? f2 : f3;
        const float* cb = fsrc + ((long)b * CCH) * (long)(H * W);
        for (int i = 0; i < 13; ++i) {
            const int bin = brow + 4 * i;
            if (bin >= NBIN) break;
            float acc[4] = {0.0f, 0.0f, 0.0f, 0.0f};
            for (int sub = 0; sub < 4; ++sub) {
                const int g = bin * 4 + sub;
                const int sp = gspat[g];
                const float w0 = gw[g * 4 + 0], w1 = gw[g * 4 + 1];
                const float w2 = gw[g * 4 + 2], w3 = gw[g * 4 + 3];
#pragma unroll
                for (int k = 0; k < 4; ++k) {
                    const float* p = cb + (long)(c4 * 4 + k) * (long)(H * W) + sp;
                    acc[k] += w0 * p[0] + w1 * p[1] + w2 * p[W] + w3 * p[W + 1];
                }
            }
#pragma unroll
            for (int k = 0; k < 4; ++k)
                st[(c4 * 4 + k) * NBIN + bin] = acc[k] * 0.25f;
        }
    }
    __syncthreads();

    // ---- phase 2: fully coalesced output write ----
    const long ob = (long)roi * OUT_PER_ROI;
#pragma unroll 7
    for (int j = 0; j < NBIN; ++j)
        outp[ob + t + 256 * j] = st[t + 256 * j];
}

// ---------------------------------------------------------------------------
// Host launcher
// ---------------------------------------------------------------------------
extern "C" void kernel_launch(void* const* d_in, const int* in_sizes, int n_in,
                              void* d_out, int out_size, void* d_ws, size_t ws_size,
                              hipStream_t stream) {
    const float* f0    = (const float*)d_in[0];
    const float* f1    = (const float*)d_in[1];
    const float* f2    = (const float*)d_in[2];
    const float* f3    = (const float*)d_in[3];
    const float* boxes = (const float*)d_in[4];
    float*       outp  = (float*)d_out;
    float*       wsf   = (float*)d_ws;

    const size_t need = (size_t)NHWC_TOTAL_FLOATS * sizeof(float);
    const bool nhwc = (ws_size >= need);

    if (nhwc) {
        const float* srcs[4] = {f0, f1, f2, f3};
        const long   offs[4] = {L0_OFF, L1_OFF, L2_OFF, L3_OFF};
        const int    hws[4]  = {200 * 200, 100 * 100, 50 * 50, 25 * 25};
        for (int L = 0; L < 4; ++L) {
            dim3 grid((hws[L] + 31) / 32, CCH / 32, BATCH);
            nchw_to_nhwc<<<grid, 32, 0, stream>>>(srcs[L], wsf + offs[L], hws[L]);
        }
    }

    roi_align<<<KROI, 256, 0, stream>>>(f0, f1, f2, f3, boxes, outp,
                                        wsf, nhwc ? 1 : 0);
}